// SinkhornOTLoss_26972394619272
// MI455X (gfx1250) — compile-verified
//
#include <hip/hip_runtime.h>

#define N 8192
#define D 128
#define EPS 0.05f
#define INV_EPS 20.0f
#define STAB 1e-8f
#define NITER 50
#define MU (1.0f / 8192.0f)      // mu == nu == 1/N
#define CCHUNKS 64               // row chunks for K^T u partial sums
#define CROWS (N / CCHUNKS)      // 128 rows per chunk

typedef __attribute__((ext_vector_type(2))) float    v2f;
typedef __attribute__((ext_vector_type(8))) float    v8f;
typedef __attribute__((ext_vector_type(8))) _Float16 v8h;

// ---------------------------------------------------------------------------
// x2[i] = sum_k feats[i][k]^2   (one thread per row, float4 loads)
// ---------------------------------------------------------------------------
__global__ void sumsq_kernel(const float* __restrict__ feats, float* __restrict__ out) {
    const int i = blockIdx.x * blockDim.x + threadIdx.x;
    const float4* f = (const float4*)(feats + (size_t)i * D);
    float s = 0.0f;
#pragma unroll
    for (int t = 0; t < D / 4; ++t) {
        const float4 q = f[t];
        s += q.x * q.x + q.y * q.y + q.z * q.z + q.w * q.w;
    }
    out[i] = s;
}

__global__ void init_ones_kernel(float* __restrict__ v) {
    v[blockIdx.x * blockDim.x + threadIdx.x] = 1.0f;
}

// ---------------------------------------------------------------------------
// Build K = exp(-max(x2_i + y2_j - 2 x_i.y_j, 0)/EPS) in fp16 via f32 WMMA.
// Wave computes a 16x64 strip: 1 A-tile (16x4 per k-step) vs 4 B-tiles.
// f32 WMMA per-lane layout (16x16x4): idx = lane%16, k = k0 + 2*(lane/16)+{0,1}
//   -> one aligned float2 load per matrix per k-step.
// C/D layout: element (m = r + 8*(lane/16), n = lane%16) in acc[r].
// grid = (N/64, N/128), block = 256 (8 waves; wave w -> row tile by*8+w)
// ---------------------------------------------------------------------------
__global__ void build_k_kernel(const float* __restrict__ src, const float* __restrict__ tgt,
                               const float* __restrict__ x2, const float* __restrict__ y2,
                               _Float16* __restrict__ Km) {
    const int lane = threadIdx.x & 31;
    const int wave = threadIdx.x >> 5;
    const int mt   = blockIdx.y * 8 + wave;   // row tile index (0..511)
    const int nt0  = blockIdx.x * 4;          // first of 4 col tiles
    const int row0 = mt * 16;
    const int mrow = row0 + (lane & 15);
    const int koff = (lane >> 4) * 2;

    v8f acc[4];
#pragma unroll
    for (int t = 0; t < 4; ++t) acc[t] = (v8f)0.0f;

    for (int k0 = 0; k0 < D; k0 += 4) {
        const int kk = k0 + koff;
        const v2f a = *(const v2f*)(src + (size_t)mrow * D + kk);
#pragma unroll
        for (int t = 0; t < 4; ++t) {
            const int ncol = (nt0 + t) * 16 + (lane & 15);
            const v2f b = *(const v2f*)(tgt + (size_t)ncol * D + kk);
            acc[t] = __builtin_amdgcn_wmma_f32_16x16x4_f32(
                false, a, false, b, (short)0, acc[t], false, false);
        }
    }

    const int jloc  = lane & 15;
    const int rhalf = (lane >> 4) * 8;
#pragma unroll
    for (int t = 0; t < 4; ++t) {
        const int j  = (nt0 + t) * 16 + jloc;
        const float yj = y2[j];
#pragma unroll
        for (int r = 0; r < 8; ++r) {
            const int i = row0 + rhalf + r;
            float cd = x2[i] + yj - 2.0f * acc[t][r];
            cd = fmaxf(cd, 0.0f);
            Km[(size_t)i * N + j] = (_Float16)__expf(-cd * INV_EPS);
        }
    }
}

// ---------------------------------------------------------------------------
// u[i] = MU / (sum_j K[i][j] * v[j] + STAB)   — one wave per row, b128 loads
// ---------------------------------------------------------------------------
__global__ void rowmv_u_kernel(const _Float16* __restrict__ Km, const float* __restrict__ v,
                               float* __restrict__ u) {
    const int lane = threadIdx.x & 31;
    const int row  = blockIdx.x * 8 + (threadIdx.x >> 5);
    const _Float16* kr = Km + (size_t)row * N;
    float s = 0.0f;
    for (int c = lane * 8; c < N; c += 256) {
        const v8h kv = *(const v8h*)(kr + c);
        const float4 a = *(const float4*)(v + c);
        const float4 b = *(const float4*)(v + c + 4);
        s += (float)kv[0] * a.x + (float)kv[1] * a.y + (float)kv[2] * a.z + (float)kv[3] * a.w
           + (float)kv[4] * b.x + (float)kv[5] * b.y + (float)kv[6] * b.z + (float)kv[7] * b.w;
    }
#pragma unroll
    for (int off = 16; off > 0; off >>= 1) s += __shfl_down(s, off, 32);
    if (lane == 0) u[row] = MU / (s + STAB);
}

// ---------------------------------------------------------------------------
// partial[chunk][j] = sum_{i in chunk} K[i][j] * u[i]
// thread owns 8 consecutive cols (16B coalesced loads); deterministic (no atomics)
// grid = (N/2048, CCHUNKS), block = 256
// ---------------------------------------------------------------------------
__global__ void colmv_partial_kernel(const _Float16* __restrict__ Km, const float* __restrict__ u,
                                     float* __restrict__ partial) {
    const int j0 = (blockIdx.x * blockDim.x + threadIdx.x) * 8;
    const int i0 = blockIdx.y * CROWS;
    float acc[8];
#pragma unroll
    for (int t = 0; t < 8; ++t) acc[t] = 0.0f;
    for (int i = i0; i < i0 + CROWS; ++i) {
        const v8h kv = *(const v8h*)(Km + (size_t)i * N + j0);
        const float ui = u[i];
#pragma unroll
        for (int t = 0; t < 8; ++t) acc[t] += (float)kv[t] * ui;
    }
    float* p = partial + (size_t)blockIdx.y * N + j0;
    *(float4*)(p)     = make_float4(acc[0], acc[1], acc[2], acc[3]);
    *(float4*)(p + 4) = make_float4(acc[4], acc[5], acc[6], acc[7]);
}

// v[j] = MU / (sum_chunk partial[chunk][j] + STAB)   (nu == mu)
__global__ void finalize_v_kernel(const float* __restrict__ partial, float* __restrict__ v) {
    const int j = blockIdx.x * blockDim.x + threadIdx.x;
    float s = 0.0f;
#pragma unroll
    for (int c = 0; c < CCHUNKS; ++c) s += partial[(size_t)c * N + j];
    v[j] = MU / (s + STAB);
}

// ---------------------------------------------------------------------------
// rowloss[i] = u[i] * sum_j K_ij * (-EPS*log(K_ij)) * v[j]
// (C recovered from the stored fp16 K for self-consistency)
// ---------------------------------------------------------------------------
__global__ void loss_rows_kernel(const _Float16* __restrict__ Km, const float* __restrict__ u,
                                 const float* __restrict__ v, float* __restrict__ rowloss) {
    const int lane = threadIdx.x & 31;
    const int row  = blockIdx.x * 8 + (threadIdx.x >> 5);
    const _Float16* kr = Km + (size_t)row * N;
    float s = 0.0f;
    for (int c = lane * 8; c < N; c += 256) {
        const v8h kv = *(const v8h*)(kr + c);
#pragma unroll
        for (int t = 0; t < 8; ++t) {
            const float kf = (float)kv[t];
            const float cc = -EPS * __logf(kf);
            s += kf * cc * v[c + t];
        }
    }
#pragma unroll
    for (int off = 16; off > 0; off >>= 1) s += __shfl_down(s, off, 32);
    if (lane == 0) rowloss[row] = u[row] * s;
}

__global__ void reduce_loss_kernel(const float* __restrict__ rowloss, float* __restrict__ out) {
    __shared__ float sm[256];
    float s = 0.0f;
    for (int i = threadIdx.x; i < N; i += 256) s += rowloss[i];
    sm[threadIdx.x] = s;
    __syncthreads();
    for (int off = 128; off > 0; off >>= 1) {
        if ((int)threadIdx.x < off) sm[threadIdx.x] += sm[threadIdx.x + off];
        __syncthreads();
    }
    if (threadIdx.x == 0) out[0] = sm[0];
}

// ---------------------------------------------------------------------------
extern "C" void kernel_launch(void* const* d_in, const int* in_sizes, int n_in,
                              void* d_out, int out_size, void* d_ws, size_t ws_size,
                              hipStream_t stream) {
    (void)in_sizes; (void)n_in; (void)out_size; (void)ws_size;
    const float* src = (const float*)d_in[0];
    const float* tgt = (const float*)d_in[1];
    float* out = (float*)d_out;

    // workspace layout (all 16B-aligned)
    char* ws = (char*)d_ws;
    _Float16* Km     = (_Float16*)ws;                          // 8192*8192*2 = 128 MB
    float*    x2     = (float*)(ws + (size_t)N * N * 2);       // 32 KB
    float*    y2     = x2 + N;                                 // 32 KB
    float*    u      = y2 + N;                                 // 32 KB
    float*    v      = u + N;                                  // 32 KB
    float*    part   = v + N;                                  // 64*8192*4 = 2 MB
    float*    rloss  = part + (size_t)CCHUNKS * N;             // 32 KB

    sumsq_kernel<<<N / 256, 256, 0, stream>>>(src, x2);
    sumsq_kernel<<<N / 256, 256, 0, stream>>>(tgt, y2);
    init_ones_kernel<<<N / 256, 256, 0, stream>>>(v);

    build_k_kernel<<<dim3(N / 64, N / 128), 256, 0, stream>>>(src, tgt, x2, y2, Km);

    for (int it = 0; it < NITER; ++it) {
        rowmv_u_kernel<<<N / 8, 256, 0, stream>>>(Km, v, u);
        colmv_partial_kernel<<<dim3(N / 2048, CCHUNKS), 256, 0, stream>>>(Km, u, part);
        finalize_v_kernel<<<N / 256, 256, 0, stream>>>(part, v);
    }

    loss_rows_kernel<<<N / 8, 256, 0, stream>>>(Km, u, v, rloss);
    reduce_loss_kernel<<<1, 256, 0, stream>>>(rloss, out);
}